// SimpleGCNLayer_33371895890435
// MI455X (gfx1250) — compile-verified
//
#include <hip/hip_runtime.h>
#include <math.h>

#define B_ 4
#define N_ 512
#define D_ 128
#define LN_EPS 1e-5f

typedef __attribute__((ext_vector_type(2))) float v2f;
typedef __attribute__((ext_vector_type(4))) float v4f;
typedef __attribute__((ext_vector_type(8))) float v8f;

// ---------------------------------------------------------------------------
// Kernel 1: stream e once: copy to output (non-temporal) and compute
// logits[b,n,m] = sqrt(sum_d e^2). One wave32 per 128-float row (16B/lane).
// This kernel is the whole roofline: ~1 GB of HBM traffic.
// ---------------------------------------------------------------------------
__global__ void k_rownorm_copy(const float* __restrict__ e,
                               float* __restrict__ e_out,
                               float* __restrict__ logits) {
  const int wave = threadIdx.x >> 5;
  const int lane = threadIdx.x & 31;
  const long row  = (long)blockIdx.x * 8 + wave;   // row < B*N*N
  const long base = row * D_ + lane * 4;
  v4f v = __builtin_nontemporal_load((const v4f*)(e + base));
  __builtin_nontemporal_store(v, (v4f*)(e_out + base));
  float s = v.x * v.x + v.y * v.y + v.z * v.z + v.w * v.w;
#pragma unroll
  for (int off = 16; off > 0; off >>= 1)
    s += __shfl_xor(s, off, 32);
  if (lane == 0) logits[row] = sqrtf(s);
}

// ---------------------------------------------------------------------------
// Kernel 2: softmax over the neighbor axis. One 256-thread block per row of
// 512 logits (in-place in the workspace buffer).
// ---------------------------------------------------------------------------
__global__ void k_softmax(float* __restrict__ w) {
  __shared__ float red[256];
  float* p = w + (long)blockIdx.x * N_;
  const int t = threadIdx.x;
  float a0 = p[t], a1 = p[t + 256];

  red[t] = fmaxf(a0, a1);
  __syncthreads();
  for (int s = 128; s > 0; s >>= 1) {
    if (t < s) red[t] = fmaxf(red[t], red[t + s]);
    __syncthreads();
  }
  const float m = red[0];
  __syncthreads();

  float e0 = expf(a0 - m), e1 = expf(a1 - m);
  red[t] = e0 + e1;
  __syncthreads();
  for (int s = 128; s > 0; s >>= 1) {
    if (t < s) red[t] += red[t + s];
    __syncthreads();
  }
  const float inv = 1.0f / red[0];
  p[t] = e0 * inv;
  p[t + 256] = e1 * inv;
}

// ---------------------------------------------------------------------------
// Kernel 3: agg = w @ h (per batch 512x512 @ 512x128), then hs = h + agg.
// Exact fp32 via V_WMMA_F32_16X16X4_F32. Block = 8 waves; each wave owns one
// 16(n) x 16(d) output tile; 8 waves cover d=128. grid = (N/16, B).
// Per-lane layout (ISA 7.12.2): M = lane&15, koff = lane<16 ? 0 : 2.
// ---------------------------------------------------------------------------
__global__ void k_agg(const float* __restrict__ w,   // [B,N,N] softmax weights
                      const float* __restrict__ h,   // [B,N,D]
                      float* __restrict__ hs) {      // [B,N,D] = h + agg
  const int wave = threadIdx.x >> 5;
  const int lane = threadIdx.x & 31;
  const int n0 = blockIdx.x * 16;
  const int b  = blockIdx.y;
  const int d0 = wave * 16;
  const int M    = lane & 15;
  const int koff = (lane >> 4) << 1;                 // 0 or 2

  const float* wrow  = w + ((long)b * N_ + (n0 + M)) * N_;
  const float* hbase = h + (long)b * N_ * D_ + d0 + M;   // h[b][*][d0+M]

  v8f acc = {};
  for (int k = 0; k < N_; k += 4) {
    v2f a, bb;
    a.x  = wrow[k + koff];
    a.y  = wrow[k + koff + 1];
    bb.x = hbase[(long)(k + koff) * D_];
    bb.y = hbase[(long)(k + koff + 1) * D_];
    acc = __builtin_amdgcn_wmma_f32_16x16x4_f32(
        false, a, false, bb, (short)0, acc, false, false);
  }

  const int col   = d0 + M;
  const int rbase = (lane < 16) ? 0 : 8;
#pragma unroll
  for (int r = 0; r < 8; ++r) {
    const long idx = ((long)b * N_ + (n0 + rbase + r)) * D_ + col;
    hs[idx] = h[idx] + acc[r];
  }
}

// ---------------------------------------------------------------------------
// Kernel 4: x = hs @ W^T + bias, LayerNorm over D, ReLU.
// Block = 8 waves covers a full 16(n) x 128(k) row-block so LN can be done in
// LDS. grid = B*N/16.
// ---------------------------------------------------------------------------
__global__ void k_linear_ln(const float* __restrict__ hs,   // [B,N,D]
                            const float* __restrict__ Wm,   // [D,D], W[k][d]
                            const float* __restrict__ bias,
                            const float* __restrict__ gamma,
                            const float* __restrict__ beta,
                            float* __restrict__ out) {      // [B,N,D]
  __shared__ float xs[16][D_];
  __shared__ float mu_s[16], rs_s[16];

  const int wave = threadIdx.x >> 5;
  const int lane = threadIdx.x & 31;
  const int b    = blockIdx.x / (N_ / 16);
  const int n0   = (blockIdx.x % (N_ / 16)) * 16;
  const int k0   = wave * 16;
  const int M    = lane & 15;
  const int koff = (lane >> 4) << 1;

  const float* arow = hs + ((long)b * N_ + (n0 + M)) * D_;  // A: rows n, cols d
  const float* wcol = Wm + (long)(k0 + M) * D_;             // B[d][k] = W[k][d]

  v8f acc = {};
  for (int d = 0; d < D_; d += 4) {
    v2f a, bb;
    a.x  = arow[d + koff];
    a.y  = arow[d + koff + 1];
    bb.x = wcol[d + koff];
    bb.y = wcol[d + koff + 1];
    acc = __builtin_amdgcn_wmma_f32_16x16x4_f32(
        false, a, false, bb, (short)0, acc, false, false);
  }

  const float bv  = bias[k0 + M];
  const int rbase = (lane < 16) ? 0 : 8;
#pragma unroll
  for (int r = 0; r < 8; ++r)
    xs[rbase + r][k0 + M] = acc[r] + bv;
  __syncthreads();

  if (threadIdx.x < 16) {
    const int row = threadIdx.x;
    float s = 0.f;
    for (int c = 0; c < D_; ++c) s += xs[row][c];
    const float mu = s * (1.0f / D_);
    float v = 0.f;
    for (int c = 0; c < D_; ++c) {
      const float d = xs[row][c] - mu;
      v += d * d;
    }
    mu_s[row] = mu;
    rs_s[row] = rsqrtf(v * (1.0f / D_) + LN_EPS);
  }
  __syncthreads();

  for (int i = threadIdx.x; i < 16 * D_; i += 256) {
    const int row = i >> 7, col = i & (D_ - 1);
    float y = (xs[row][col] - mu_s[row]) * rs_s[row] * gamma[col] + beta[col];
    out[((long)b * N_ + (n0 + row)) * D_ + col] = fmaxf(y, 0.f);
  }
}

// ---------------------------------------------------------------------------
extern "C" void kernel_launch(void* const* d_in, const int* in_sizes, int n_in,
                              void* d_out, int out_size, void* d_ws, size_t ws_size,
                              hipStream_t stream) {
  const float* h     = (const float*)d_in[0];
  const float* e     = (const float*)d_in[1];
  const float* Wm    = (const float*)d_in[2];
  const float* bias  = (const float*)d_in[3];
  const float* gamma = (const float*)d_in[4];
  const float* beta  = (const float*)d_in[5];

  float* out_h = (float*)d_out;                       // [B,N,D]
  float* out_e = out_h + (long)B_ * N_ * D_;          // e copy (tuple 2nd elem)

  float* w_buf  = (float*)d_ws;                       // [B,N,N] logits -> weights
  float* hs_buf = w_buf + (long)B_ * N_ * N_;         // [B,N,D] h + agg

  // 1) stream e: copy out + row norms (the only roofline-relevant kernel)
  k_rownorm_copy<<<(B_ * N_ * N_) / 8, 256, 0, stream>>>(e, out_e, w_buf);
  // 2) softmax over neighbors, in place
  k_softmax<<<B_ * N_, 256, 0, stream>>>(w_buf);
  // 3) agg = w @ h ; hs = h + agg   (fp32 WMMA)
  k_agg<<<dim3(N_ / 16, B_), 256, 0, stream>>>(w_buf, h, hs_buf);
  // 4) Linear + LayerNorm + ReLU     (fp32 WMMA)
  k_linear_ln<<<B_ * N_ / 16, 256, 0, stream>>>(hs_buf, Wm, bias, gamma, beta, out_h);
}